// FixedDGCNNBackbone_81037442941004
// MI455X (gfx1250) — compile-verified
//
#include <hip/hip_runtime.h>
#include <hip/hip_bf16.h>

typedef __attribute__((ext_vector_type(2))) float v2f;
typedef __attribute__((ext_vector_type(8))) float v8f;

__device__ __forceinline__ v8f wmma4(v2f a, v2f b, v8f c) {
  // D(16x16,f32) = A(16x4,f32) * B(4x16,f32) + C
  return __builtin_amdgcn_wmma_f32_16x16x4_f32(false, a, false, b, (short)0, c, false, false);
}

#define NPT 4096
#define KNN 20
#define NEGS 0.2f

// ---------------------------------------------------------------------------
// Per-point squared norms (feeds the WMMA k-NN).
// ---------------------------------------------------------------------------
__global__ void sq_kernel(const float* __restrict__ x, int stride, int C,
                          float* __restrict__ sqout) {
  int i = blockIdx.x * blockDim.x + threadIdx.x;
  if (i >= 2 * NPT) return;
  const float* r = x + (size_t)i * stride;
  float s = 0.f;
  for (int c = 0; c < C; ++c) { float v = r[c]; s += v * v; }
  sqout[i] = s;
}

// ---------------------------------------------------------------------------
// WMMA k-NN: pd = 2*X.X^T - sq - sq^T on the f32 matrix pipe.
// Block = 128 threads (4 waves), 64 queries/block (16 per wave, as B cols).
// 16-candidate tiles (A rows) staged in LDS, shared by all 4 waves.
// Per-lane top-20 kept UNSORTED with tracked (minv, minpos): accept test is
// pd > minv; replacement is a flat j==minpos select chain + 19-compare rescan
// (no array rotation -> no register-copy walls). Lane pair (L, L+16) merges
// via a 20-round shfl_xor(16) argmax.
// ---------------------------------------------------------------------------
template<int C>
__global__ void __launch_bounds__(128) knn_kernel(const float* __restrict__ x,
                                                  int stride,
                                                  const float* __restrict__ sqg,
                                                  int* __restrict__ idx_out) {
  __shared__ float Q[64][C + 4];
  __shared__ float T[16][C + 4];
  __shared__ float Tsq[16];
  const int tid = threadIdx.x, lane = tid & 31, wave = tid >> 5;
  const int b = blockIdx.y;
  const int qb = blockIdx.x * 64;            // block's first query
  const int qn = lane & 15;                  // query within wave tile
  const int q0 = qb + wave * 16;             // wave's first query
  const int kb = 2 * (lane >> 4);
  const float* xb = x + (size_t)b * NPT * stride;
  const float* sqb = sqg + (size_t)b * NPT;

  for (int t = tid; t < 64 * C; t += 128) {
    int r = t / C, c = t % C;
    Q[r][c] = xb[(size_t)(qb + r) * stride + c];
  }
  __syncthreads();
  const float sqq = sqb[q0 + qn];

  float bd[KNN]; int bi[KNN];
#pragma unroll
  for (int j = 0; j < KNN; ++j) { bd[j] = -3.4e38f; bi[j] = -1; }
  float minv = -3.4e38f;
  int minpos = 0;

#pragma unroll 1
  for (int m0 = 0; m0 < NPT; m0 += 16) {
    __syncthreads();
    for (int t = tid; t < 16 * C; t += 128) {
      int r = t / C, c = t % C;
      T[r][c] = xb[(size_t)(m0 + r) * stride + c];
    }
    if (tid < 16) Tsq[tid] = sqb[m0 + tid];
    __syncthreads();

    v8f acc = {};
    for (int k4 = 0; k4 < C; k4 += 4) {
      v2f a, bf;
      a.x  = T[lane & 15][k4 + kb];        a.y  = T[lane & 15][k4 + kb + 1];
      bf.x = Q[wave * 16 + qn][k4 + kb];   bf.y = Q[wave * 16 + qn][k4 + kb + 1];
      acc = wmma4(a, bf, acc);
    }

    float pdv[8];
#pragma unroll
    for (int v = 0; v < 8; ++v)
      pdv[v] = 2.f * acc[v] - sqq - Tsq[v + 8 * (lane >> 4)];  // larger == closer
    float mx8 = pdv[0];
#pragma unroll
    for (int v = 1; v < 8; ++v) mx8 = fmaxf(mx8, pdv[v]);

    if (mx8 > minv) {                 // rare: any candidate beats current min
#pragma unroll
      for (int v = 0; v < 8; ++v) {
        float pd = pdv[v];
        if (pd > minv) {
          int ic = m0 + v + 8 * (lane >> 4);
#pragma unroll
          for (int j = 0; j < KNN; ++j) {    // flat select chain, no rotation
            bd[j] = (j == minpos) ? pd : bd[j];
            bi[j] = (j == minpos) ? ic : bi[j];
          }
          minv = bd[0]; minpos = 0;
#pragma unroll
          for (int j = 1; j < KNN; ++j)
            if (bd[j] < minv) { minv = bd[j]; minpos = j; }
        }
      }
    }
  }

  // merge lane pair (L, L+16) -> top-20 for query q0+qn (lists unsorted)
  int* outp = idx_out + ((size_t)b * NPT + q0 + qn) * KNN;
#pragma unroll 1
  for (int r = 0; r < KNN; ++r) {
    float md = bd[0]; int mi = bi[0]; int mj = 0;
#pragma unroll
    for (int j = 1; j < KNN; ++j)
      if (bd[j] > md) { md = bd[j]; mi = bi[j]; mj = j; }
    float od = __shfl_xor(md, 16);
    int oi = __shfl_xor(mi, 16);
    bool takeOther = (lane < 16) ? (od > md) : (od >= md);  // lower lane wins ties
    if (!takeOther) {       // this lane's element won: pop it
#pragma unroll
      for (int j = 0; j < KNN; ++j) bd[j] = (j == mj) ? -3.4e38f : bd[j];
    } else { md = od; mi = oi; }
    if (lane < 16) outp[r] = mi;
  }
}

// ---------------------------------------------------------------------------
// Edge conv: one block per point. Edges staged to LDS, WMMA f32 over two
// M-tiles (k=0..15, k=16..31 w/ rows >=20 replicated from edge 0: max-safe,
// masked out of BN sums). Outputs max_k h (pre-BN) + atomic channel stats.
// ---------------------------------------------------------------------------
template<int CIN, int COUT>
__global__ void __launch_bounds__(128) edge_conv_kernel(
    const float* __restrict__ xin, int in_stride,
    const int* __restrict__ idx,
    const float* __restrict__ W,       // (COUT, 2*CIN) row-major
    float* __restrict__ out, int out_stride,
    float* __restrict__ stats) {       // [COUT] sum, [COUT] sumsq
  constexpr int C2 = 2 * CIN;
  __shared__ float E[KNN][C2 + 4];
  const int tid = threadIdx.x, lane = tid & 31, wave = tid >> 5;
  const int b = blockIdx.y, n = blockIdx.x;
  const float* xb = xin + (size_t)b * NPT * in_stride;
  const float* xc = xb + (size_t)n * in_stride;
  const int* nb = idx + ((size_t)b * NPT + n) * KNN;

  for (int t = tid; t < KNN * CIN; t += 128) {
    int k = t / CIN, c = t % CIN;
    float cv = xc[c];
    float nv = xb[(size_t)nb[k] * in_stride + c];
    E[k][c] = nv - cv;          // nbr - center
    E[k][CIN + c] = cv;         // center
  }
  __syncthreads();

  const int mrow = lane & 15;
  const int kb = 2 * (lane >> 4);
  const int r0 = mrow;
  const int r1e = 16 + mrow;
  const int r1 = (r1e < KNN) ? r1e : 0;   // replicate edge 0 in padded rows

  for (int nc = wave; nc < COUT / 16; nc += 4) {
    const int ocol = nc * 16 + mrow;
    const float* Wr = W + (size_t)ocol * C2;
    v8f acc0 = {}; v8f acc1 = {};
    for (int k4 = 0; k4 < C2; k4 += 4) {
      v2f a0, a1, bf;
      a0.x = E[r0][k4 + kb]; a0.y = E[r0][k4 + kb + 1];
      a1.x = E[r1][k4 + kb]; a1.y = E[r1][k4 + kb + 1];
      bf.x = Wr[k4 + kb];    bf.y = Wr[k4 + kb + 1];
      acc0 = wmma4(a0, bf, acc0);
      acc1 = wmma4(a1, bf, acc1);
    }
    float mx = acc0[0], s = 0.f, s2 = 0.f;
#pragma unroll
    for (int v = 0; v < 8; ++v) { float h = acc0[v]; mx = fmaxf(mx, h); s += h; s2 += h * h; }
#pragma unroll
    for (int v = 0; v < 8; ++v) mx = fmaxf(mx, acc1[v]);   // dup rows: max-safe
    if (lane < 16) {   // valid tile-1 rows: M=0..3 (k=16..19), lanes 0..15
#pragma unroll
      for (int v = 0; v < 4; ++v) { float h = acc1[v]; s += h; s2 += h * h; }
    }
    mx = fmaxf(mx, __shfl_xor(mx, 16));
    s += __shfl_xor(s, 16);
    s2 += __shfl_xor(s2, 16);
    if (lane < 16) {
      out[((size_t)b * NPT + n) * out_stride + ocol] = mx;
      atomicAdd(&stats[ocol], s);
      atomicAdd(&stats[COUT + ocol], s2);
    }
  }
}

// ---------------------------------------------------------------------------
// Generic WMMA f32 GEMM: Y(M x N) = X(M x KD) * W(N x KD)^T + bias, with
// optional per-column BN stat accumulation. Block: 16 rows x 64 cols (4 waves).
// ---------------------------------------------------------------------------
template<int KD>
__global__ void __launch_bounds__(128) gemm_kernel(
    const float* __restrict__ X, int xstride,
    const float* __restrict__ W,
    const float* __restrict__ bias,
    float* __restrict__ Y, int ystride,
    float* __restrict__ stats, int Ncols) {
  __shared__ float Xt[16][KD + 4];
  const int tid = threadIdx.x, lane = tid & 31, wave = tid >> 5;
  const int m0 = blockIdx.x * 16;
  for (int t = tid; t < 16 * KD; t += 128) {
    int r = t / KD, c = t % KD;
    Xt[r][c] = X[(size_t)(m0 + r) * xstride + c];
  }
  __syncthreads();
  const int col = blockIdx.y * 64 + wave * 16 + (lane & 15);
  const bool colok = col < Ncols;
  const int kb = 2 * (lane >> 4);
  const int ar = lane & 15;
  const float* Wr = W + (size_t)(colok ? col : 0) * KD;
  v8f acc = {};
  for (int k4 = 0; k4 < KD; k4 += 4) {
    v2f a, bf;
    a.x = Xt[ar][k4 + kb]; a.y = Xt[ar][k4 + kb + 1];
    float w0 = Wr[k4 + kb], w1 = Wr[k4 + kb + 1];
    bf.x = colok ? w0 : 0.f; bf.y = colok ? w1 : 0.f;
    acc = wmma4(a, bf, acc);
  }
  const float bv = (bias != nullptr && colok) ? bias[col] : 0.f;
  float s = 0.f, s2 = 0.f;
#pragma unroll
  for (int v = 0; v < 8; ++v) {
    float h = acc[v] + bv;
    int m = m0 + v + 8 * (lane >> 4);
    if (colok) Y[(size_t)m * ystride + col] = h;
    s += h; s2 += h * h;
  }
  if (stats != nullptr) {
    s += __shfl_xor(s, 16);
    s2 += __shfl_xor(s2, 16);
    if (lane < 16 && colok) {
      atomicAdd(&stats[col], s);
      atomicAdd(&stats[Ncols + col], s2);
    }
  }
}

// ---------------------------------------------------------------------------
// In-place BN (+optional leaky) finalize from accumulated (sum, sumsq).
// ---------------------------------------------------------------------------
__global__ void bn_finalize_kernel(float* __restrict__ y, int stride,
                                   const float* __restrict__ stats,
                                   const float* __restrict__ gamma,
                                   const float* __restrict__ beta,
                                   int C, int rows, float inv_count, int leaky) {
  int i = blockIdx.x * blockDim.x + threadIdx.x;
  if (i >= rows * C) return;
  int c = i % C, r = i / C;
  float mu = stats[c] * inv_count;
  float var = stats[C + c] * inv_count - mu * mu;
  float sc = rsqrtf(var + 1e-5f) * gamma[c];
  float v = y[(size_t)r * stride + c];
  float o = (v - mu) * sc + beta[c];
  if (leaky) o = (o >= 0.f) ? o : NEGS * o;
  y[(size_t)r * stride + c] = o;
}

__global__ void zero_kernel(float* __restrict__ p, int n) {
  int i = blockIdx.x * blockDim.x + threadIdx.x;
  if (i < n) p[i] = 0.f;
}

// ---------------------------------------------------------------------------
extern "C" void kernel_launch(void* const* d_in, const int* in_sizes, int n_in,
                              void* d_out, int out_size, void* d_ws, size_t ws_size,
                              hipStream_t stream) {
  (void)in_sizes; (void)n_in; (void)out_size; (void)ws_size;
  const float* x    = (const float*)d_in[0];
  const float* W1   = (const float*)d_in[1];
  const float* g1   = (const float*)d_in[2];
  const float* b1   = (const float*)d_in[3];
  const float* W2   = (const float*)d_in[4];
  const float* g2   = (const float*)d_in[5];
  const float* b2   = (const float*)d_in[6];
  const float* W3   = (const float*)d_in[7];
  const float* g3   = (const float*)d_in[8];
  const float* b3   = (const float*)d_in[9];
  const float* W4   = (const float*)d_in[10];
  const float* g4   = (const float*)d_in[11];
  const float* b4   = (const float*)d_in[12];
  const float* W5   = (const float*)d_in[13];
  const float* g5   = (const float*)d_in[14];
  const float* b5   = (const float*)d_in[15];
  const float* Fw[4] = {(const float*)d_in[16], (const float*)d_in[20],
                        (const float*)d_in[24], (const float*)d_in[28]};
  const float* Fb[4] = {(const float*)d_in[17], (const float*)d_in[21],
                        (const float*)d_in[25], (const float*)d_in[29]};
  const float* gn[4] = {(const float*)d_in[18], (const float*)d_in[22],
                        (const float*)d_in[26], (const float*)d_in[30]};
  const float* bb[4] = {(const float*)d_in[19], (const float*)d_in[23],
                        (const float*)d_in[27], (const float*)d_in[31]};
  const float* semW = (const float*)d_in[32];
  const float* semb = (const float*)d_in[33];

  const size_t BN_ = 2 * (size_t)NPT;          // 8192 points
  float* ws    = (float*)d_ws;
  float* xcat  = ws;                           // 8192 x 512
  float* h5    = ws + BN_ * 512;               // 8192 x 512
  float* stats = ws + 2 * BN_ * 512;           // 9 slots x 1024
  float* sqg   = ws + 2 * BN_ * 512 + 9 * 1024;        // 8192 norms
  int*   idx   = (int*)(ws + 2 * BN_ * 512 + 9 * 1024 + BN_);
  float* out   = (float*)d_out;

  const float invE = 1.f / (float)(BN_ * KNN); // edge-conv BN count
  const float invP = 1.f / (float)BN_;         // point BN count

  zero_kernel<<<(9 * 1024 + 255) / 256, 256, 0, stream>>>(stats, 9 * 1024);

  dim3 kgrid(NPT / 64, 2);                     // WMMA k-NN: 64 queries/block
  dim3 egrid(NPT, 2);
  const int sqblocks = (int)(BN_ + 255) / 256;
  auto fin = [&](float* y, float* st, const float* g, const float* b,
                 int C, int leaky, float inv) {
    int tot = (int)BN_ * C;
    bn_finalize_kernel<<<(tot + 255) / 256, 256, 0, stream>>>(y, 512, st, g, b, C, (int)BN_, inv, leaky);
  };

  // layer 1: x(C=4) -> xcat[0:64)
  sq_kernel<<<sqblocks, 256, 0, stream>>>(x, 4, 4, sqg);
  knn_kernel<4><<<kgrid, 128, 0, stream>>>(x, 4, sqg, idx);
  edge_conv_kernel<4, 64><<<egrid, 128, 0, stream>>>(x, 4, idx, W1, xcat, 512, stats);
  fin(xcat, stats, g1, b1, 64, 1, invE);
  // layer 2: x1 -> xcat[64:128)
  sq_kernel<<<sqblocks, 256, 0, stream>>>(xcat, 512, 64, sqg);
  knn_kernel<64><<<kgrid, 128, 0, stream>>>(xcat, 512, sqg, idx);
  edge_conv_kernel<64, 64><<<egrid, 128, 0, stream>>>(xcat, 512, idx, W2, xcat + 64, 512, stats + 1024);
  fin(xcat + 64, stats + 1024, g2, b2, 64, 1, invE);
  // layer 3: x2 -> xcat[128:256)
  sq_kernel<<<sqblocks, 256, 0, stream>>>(xcat + 64, 512, 64, sqg);
  knn_kernel<64><<<kgrid, 128, 0, stream>>>(xcat + 64, 512, sqg, idx);
  edge_conv_kernel<64, 128><<<egrid, 128, 0, stream>>>(xcat + 64, 512, idx, W3, xcat + 128, 512, stats + 2048);
  fin(xcat + 128, stats + 2048, g3, b3, 128, 1, invE);
  // layer 4: x3 -> xcat[256:512)
  sq_kernel<<<sqblocks, 256, 0, stream>>>(xcat + 128, 512, 128, sqg);
  knn_kernel<128><<<kgrid, 128, 0, stream>>>(xcat + 128, 512, sqg, idx);
  edge_conv_kernel<128, 256><<<egrid, 128, 0, stream>>>(xcat + 128, 512, idx, W4, xcat + 256, 512, stats + 3072);
  fin(xcat + 256, stats + 3072, g4, b4, 256, 1, invE);

  // h5 = leaky(bn(xcat @ W5^T))
  dim3 g5grid(BN_ / 16, 8);
  gemm_kernel<512><<<g5grid, 128, 0, stream>>>(xcat, 512, W5, nullptr, h5, 512, stats + 4096, 512);
  fin(h5, stats + 4096, g5, b5, 512, 1, invP);

  // FC heads -> d_out (BN in place, no leaky)
  for (int i = 0; i < 4; ++i) {
    float* fi = out + (size_t)i * BN_ * 256;
    float* st = stats + (5 + i) * 1024;
    dim3 fgrid(BN_ / 16, 4);
    gemm_kernel<512><<<fgrid, 128, 0, stream>>>(h5, 512, Fw[i], Fb[i], fi, 256, st, 256);
    int tot = (int)BN_ * 256;
    bn_finalize_kernel<<<(tot + 255) / 256, 256, 0, stream>>>(fi, 256, st, gn[i], bb[i], 256, (int)BN_, invP, 0);
  }

  // sem_logits = feats[3] @ semW^T + semb  (N=20, column-masked)
  dim3 sgrid(BN_ / 16, 1);
  gemm_kernel<256><<<sgrid, 128, 0, stream>>>(out + 3 * BN_ * 256, 256, semW, semb,
                                              out + 4 * BN_ * 256, 20, nullptr, 20);
}